// FavorAttention_65695819760152
// MI455X (gfx1250) — compile-verified
//
#include <hip/hip_runtime.h>
#include <hip/hip_bf16.h>

// FAVOR-style attention, restructured:
//   s_t = temp^2 * q_t^T (feat feat^T) k_t  ;  out = cumsum(s*v)/cumsum(s)
// Pipeline: A) M = temp^2 * feat@feat^T (64x64, symmetric)
//           B) per 64-pos tile: KM = K@M via v_wmma_f32_16x16x4_f32,
//              s_t = q_t . KM_t ; tile aggregates sumS, sumSV[d]
//           C) exclusive scan of tile aggregates over tiles (per batch)
//           D) per-tile outputs using carried prefixes (fully parallel)
//
// Bandwidth-bound problem (~82 MB of mandatory traffic -> ~3.5 us at 23.3 TB/s),
// so compute stays fp32 on the native V_WMMA_F32_16X16X4_F32 path.

#define BB   32
#define SS   2048
#define DD   64
#define FF   256
#define CH   64            // positions per tile/chunk
#define NCH  (SS / CH)     // 32 chunks per batch
#define TEMP2 0.125f       // (D^-0.25)^2 = 64^-0.5

typedef float v2f __attribute__((ext_vector_type(2)));
typedef float v8f __attribute__((ext_vector_type(8)));

// ---- workspace layout (floats) ----
#define OFF_M    0                        // 64*64
#define OFF_S    4096                     // B*S
#define OFF_CS   (OFF_S   + BB * SS)      // B*NCH  (chunk sums of s)
#define OFF_PS   (OFF_CS  + BB * NCH)     // B*NCH  (exclusive prefix of chunk sums)
#define OFF_CSV  (OFF_PS  + BB * NCH)     // B*NCH*D (chunk sums of s*v)
#define OFF_PSV  (OFF_CSV + BB * NCH * DD)// B*NCH*D (exclusive prefix)

// ---------------- Kernel A: M = temp^2 * feat @ feat^T ----------------
__global__ void favor_build_M(const float* __restrict__ feat, float* __restrict__ M) {
    int idx = blockIdx.x * 256 + threadIdx.x;   // 4096 outputs, grid = 16 x 256
    int i = idx >> 6, j = idx & 63;
    const float* fi = feat + i * FF;
    const float* fj = feat + j * FF;
    float acc = 0.f;
#pragma unroll 8
    for (int f = 0; f < FF; ++f) acc += fi[f] * fj[f];
    M[idx] = TEMP2 * acc;
}

// ---------------- Kernel B: per-tile s + aggregates (WMMA) ----------------
__global__ void favor_scores(const float* __restrict__ q,
                             const float* __restrict__ k,
                             const float* __restrict__ v,
                             float* __restrict__ ws) {
    const int c = blockIdx.x;          // chunk
    const int b = blockIdx.y;          // batch*head
    const int base = c * CH;           // first position of tile
    const int tid  = threadIdx.x;      // 256 threads = 8 wave32
    const int wave = tid >> 5;
    const int lane = tid & 31;

    // M staged in LDS, pair-interleaved over K so a B fragment is one
    // contiguous float2: Ms[(k>>1)*128 + 2*n + (k&1)] = M[k][n].
    __shared__ float Ms[DD * DD];      // 16 KB
    __shared__ float s_sh[CH];

    const float* Mw = ws + OFF_M;
    for (int i = tid; i < DD * DD; i += 256) {
        const int kk = i >> 6, n = i & 63;
        Ms[(kk >> 1) * 128 + 2 * n + (kk & 1)] = Mw[i];
    }
    if (tid < CH) s_sh[tid] = 0.f;
    __syncthreads();

    // Wave tiling of the 64x64 KM block: 4 row-blocks x 4 col-blocks of 16x16.
    // waves 0..3 -> cols 0..31, waves 4..7 -> cols 32..63.
    const int r   = wave & 3;          // row-block
    const int cb0 = (wave >> 2) * 32;  // first column of this wave's two tiles
    const int hi  = lane >> 4;         // 0: lanes 0-15, 1: lanes 16-31
    const int ln  = lane & 15;

    const float* krow  = k + (((size_t)b * SS + base + r * 16 + ln) * DD);
    const float* qbase = q + (((size_t)b * SS + base) * DD);

    // prefetch the q tile needed in the dot phase (global_prefetch_b8)
    __builtin_prefetch(qbase + (r * 16 + ln) * DD, 0, 0);

    v8f acc0 = {};                     // KM rows r*16.., cols cb0..cb0+15
    v8f acc1 = {};                     // KM rows r*16.., cols cb0+16..cb0+31

#pragma unroll
    for (int kk0 = 0; kk0 < DD; kk0 += 4) {
        const int kA = kk0 + 2 * hi;   // A fp32 16x4 layout: K = 2*(lane/16)+vgpr
        // single 8-byte global load (kA even -> aligned)
        v2f a = *(const v2f*)(krow + kA);
        // single ds_load_b64 per fragment from the pair-interleaved layout
        const float* brow = &Ms[(kk0 / 2 + hi) * 128];
        v2f bf0 = *(const v2f*)(brow + 2 * (cb0 + ln));
        acc0 = __builtin_amdgcn_wmma_f32_16x16x4_f32(
            false, a, false, bf0, (short)0, acc0, false, false);
        v2f bf1 = *(const v2f*)(brow + 2 * (cb0 + 16 + ln));
        acc1 = __builtin_amdgcn_wmma_f32_16x16x4_f32(
            false, a, false, bf1, (short)0, acc1, false, false);
    }

    // s_t = q_t . KM_t : C/D layout row = vgpr + 8*(lane/16), col = lane&15
#pragma unroll
    for (int j = 0; j < 8; ++j) {
        const int m = r * 16 + j + 8 * hi;    // local row (position in tile)
        float prod = qbase[m * DD + cb0 + ln]      * acc0[j]
                   + qbase[m * DD + cb0 + 16 + ln] * acc1[j];
        // reduce over the 16 lanes sharing this row (keep lane halves separate)
        prod += __shfl_xor(prod, 1, 32);
        prod += __shfl_xor(prod, 2, 32);
        prod += __shfl_xor(prod, 4, 32);
        prod += __shfl_xor(prod, 8, 32);
        if (ln == 0) atomicAdd(&s_sh[m], prod);   // ds_add_f32
    }
    __syncthreads();

    // write s and per-chunk aggregates
    float* ws_s   = ws + OFF_S  + (size_t)b * SS + base;
    float* ws_cs  = ws + OFF_CS + b * NCH + c;
    float* ws_csv = ws + OFF_CSV + ((size_t)(b * NCH + c)) * DD;

    if (tid < CH) ws_s[tid] = s_sh[tid];
    if (tid == 0) {
        float t = 0.f;
#pragma unroll
        for (int p = 0; p < CH; ++p) t += s_sh[p];
        ws_cs[0] = t;
    }
    if (tid < DD) {
        const int d = tid;
        const float* vb = v + (((size_t)b * SS + base) * DD) + d;
        float acc = 0.f;
#pragma unroll 8
        for (int p = 0; p < CH; ++p) acc += s_sh[p] * vb[p * DD];
        ws_csv[d] = acc;
    }
}

// ---------------- Kernel C: exclusive scan of chunk aggregates ----------------
__global__ void favor_scan_chunks(float* __restrict__ ws) {
    const int b = blockIdx.x;          // grid = 32, block = 64 (thread = d)
    const int d = threadIdx.x;
    float acc = 0.f;
    for (int c = 0; c < NCH; ++c) {
        const size_t i = ((size_t)(b * NCH + c)) * DD + d;
        float t = ws[OFF_CSV + i];
        ws[OFF_PSV + i] = acc;
        acc += t;
    }
    if (d == 0) {
        float a2 = 0.f;
        for (int c = 0; c < NCH; ++c) {
            float t = ws[OFF_CS + b * NCH + c];
            ws[OFF_PS + b * NCH + c] = a2;
            a2 += t;
        }
    }
}

// ---------------- Kernel D: per-tile outputs ----------------
__global__ void favor_output(const float* __restrict__ v,
                             const float* __restrict__ ws,
                             float* __restrict__ out) {
    const int c = blockIdx.x;
    const int b = blockIdx.y;
    const int base = c * CH;
    const int tid = threadIdx.x;       // 64 threads (thread = d)

    __shared__ float s_sh[CH];
    __shared__ float cums[CH];

    s_sh[tid] = ws[OFF_S + (size_t)b * SS + base + tid];
    __syncthreads();
    if (tid == 0) {
        float run = ws[OFF_PS + b * NCH + c];
#pragma unroll
        for (int p = 0; p < CH; ++p) { run += s_sh[p]; cums[p] = run; }
    }
    __syncthreads();

    const int d = tid;
    float acc = ws[OFF_PSV + ((size_t)(b * NCH + c)) * DD + d];
    const float* vb = v   + (((size_t)b * SS + base) * DD) + d;
    float*       ob = out + (((size_t)b * SS + base) * DD) + d;
#pragma unroll 4
    for (int p = 0; p < CH; ++p) {
        acc += s_sh[p] * vb[p * DD];
        ob[p * DD] = acc / cums[p];
    }
}

extern "C" void kernel_launch(void* const* d_in, const int* in_sizes, int n_in,
                              void* d_out, int out_size, void* d_ws, size_t ws_size,
                              hipStream_t stream) {
    const float* q    = (const float*)d_in[0];
    const float* k    = (const float*)d_in[1];
    const float* v    = (const float*)d_in[2];
    const float* feat = (const float*)d_in[3];
    float* out = (float*)d_out;
    float* ws  = (float*)d_ws;

    favor_build_M<<<16, 256, 0, stream>>>(feat, ws + OFF_M);
    favor_scores<<<dim3(NCH, BB), 256, 0, stream>>>(q, k, v, ws);
    favor_scan_chunks<<<BB, DD, 0, stream>>>(ws);
    favor_output<<<dim3(NCH, BB), DD, 0, stream>>>(v, ws, out);
}